// HairBundleSDE_90752658964486
// MI455X (gfx1250) — compile-verified
//
#include <hip/hip_runtime.h>
#include <cmath>

// ---------------------------------------------------------------------------
// HairBundle SDE drift (dx) + constant diagonal diffusion (g) for MI455X.
// Pure HBM-streaming workload: ~416 MB per launch -> ~18 us floor @ 23.3 TB/s.
// No matmul structure => no WMMA; optimize data movement instead:
//   * gfx1250 async DMA (global_load_async_to_lds_b128 /
//     global_store_async_from_lds_b128, ASYNCcnt) for the LDS staging —
//     no VGPR round-trip, single s_wait_asynccnt drain
//   * LDS-staged transpose so ALL global traffic is unit-stride b128
//   * nontemporal (TH=NT) stores for the streamed outputs
//   * g emitted as a constant pattern (zero reads)
// ---------------------------------------------------------------------------

typedef float v4f __attribute__((ext_vector_type(4)));
typedef int   v4i __attribute__((ext_vector_type(4)));

#define ROWS_PER_BLOCK 256
#define FLOATS_PER_BLOCK (ROWS_PER_BLOCK * 5)   // 1280
#define VEC4_PER_BLOCK   (FLOATS_PER_BLOCK / 4) // 320
#define B_TOTAL 4000000

#if defined(__has_builtin)
#if __has_builtin(__builtin_amdgcn_global_load_async_to_lds_b128) && \
    __has_builtin(__builtin_amdgcn_global_store_async_from_lds_b128)
#define HAVE_ASYNC_LDS 1
#endif
#if __has_builtin(__builtin_amdgcn_s_wait_asynccnt)
#define WAIT_ASYNC0() __builtin_amdgcn_s_wait_asynccnt(0)
#endif
#endif
#ifndef HAVE_ASYNC_LDS
#define HAVE_ASYNC_LDS 0
#endif
#ifndef WAIT_ASYNC0
#define WAIT_ASYNC0() asm volatile("s_wait_asynccnt 0x0" ::: "memory")
#endif

#if HAVE_ASYNC_LDS
// Builtin signature (from hipcc diagnostic): param0 is int4* in AS1 (global),
// param1 is the LDS-side int4* (AS3), then two immediates (offset, cpol).
typedef __attribute__((address_space(1))) v4i gbl_v4i;
typedef __attribute__((address_space(3))) v4i lds_v4i;
#define ASYNC_G2L_B128(gptr, lptr) \
    __builtin_amdgcn_global_load_async_to_lds_b128( \
        (gbl_v4i*)(gptr), (lds_v4i*)(lptr), 0, 0)
#define ASYNC_L2G_B128(gptr, lptr) \
    __builtin_amdgcn_global_store_async_from_lds_b128( \
        (gbl_v4i*)(gptr), (lds_v4i*)(lptr), 0, 0)
#endif

__global__ __launch_bounds__(ROWS_PER_BLOCK)
void hairbundle_dx_kernel(const float* __restrict__ t_ptr,
                          const float* __restrict__ x,
                          float* __restrict__ dx,
                          float cam_coef, float cags_coef,
                          float d_over_kt, float e_exp,
                          float t_scale)
{
    __shared__ v4f lds4[VEC4_PER_BLOCK];
    float* lds = (float*)lds4;

    const int tid = threadIdx.x;
    const unsigned long long blk = blockIdx.x;

    // ---- stage 256 rows (320 x b128) global -> LDS ----
    const v4f* x4 = (const v4f*)x + blk * VEC4_PER_BLOCK;
#if HAVE_ASYNC_LDS
    // gfx1250 async DMA: cache -> LDS directly, tracked by ASYNCcnt.
    ASYNC_G2L_B128(&x4[tid], &lds4[tid]);
    if (tid < VEC4_PER_BLOCK - ROWS_PER_BLOCK)
        ASYNC_G2L_B128(&x4[tid + ROWS_PER_BLOCK], &lds4[tid + ROWS_PER_BLOCK]);
    WAIT_ASYNC0();
#else
    #pragma unroll
    for (int j = 0; j < 2; ++j) {
        int idx = tid + j * ROWS_PER_BLOCK;
        if (idx < VEC4_PER_BLOCK)
            lds4[idx] = __builtin_nontemporal_load(&x4[idx]);
    }
#endif
    __syncthreads();

    // ---- per-row math (stride-5 LDS reads: gcd(5,64)=1 -> conflict-free) ----
    const float x_hb  = lds[tid * 5 + 0];
    const float x_a   = lds[tid * 5 + 1];
    const float p_m   = fminf(fmaxf(lds[tid * 5 + 2], 0.0f), 1.0f);
    const float p_gs  = fminf(fmaxf(lds[tid * 5 + 3], 0.0f), 1.0f);
    const float p_t   = fminf(fmaxf(lds[tid * 5 + 4], 0.0f), 1.0f);

    const float t     = t_ptr[0];                       // uniform scalar load
    const float k_gs  = 1.0f - 0.5f * p_gs;             // K_GS_MAX - p_gs*DELTA_KGS
    const float u     = 0.14f * x_hb - x_a + 0.2f;      // GAMMA*x_hb - x_a + X_C
    const float f_gs  = k_gs * (u - 0.5f * p_t);        // ... - p_t*D

    // force term: omega = row * OMEGA_0/HALF
    const unsigned long long row_g = blk * ROWS_PER_BLOCK + (unsigned)tid;
    const float omega = (float)row_g * t_scale;
    const float wt    = omega * t;
    const float force = -__sinf(wt) + 0.1f * omega * __cosf(wt);

    // dx_hb = -(GAMMA*N*f_gs + K_SP*x_hb)/LAMBDA_HB + force/LAMBDA_HB
    const float dx_hb = -(7.0f * f_gs + 0.6f * x_hb) + force;

    const float c     = 0.06f - 0.06f * p_m;            // C_MAX - p_m*DELTA_C
    const float s     = 0.1f  + 0.9f  * p_m;            // S_MIN + p_m*DELTA_S
    const float relu  = fmaxf(f_gs, 0.0f);              // (f+|f|)/2
    const float dx_a  = -c + s * (relu - 0.5f * x_a);   // K_ES*(x_a - X_ES)

    const float ca_m  = p_t * cam_coef;                 // -i_t_ca/(CA2_DENOM*R_M)
    const float ca_gs = p_t * cags_coef;
    const float dp_m  = 5.0f * ca_m  * (1.0f - p_m)  - 2.0f * p_m;
    const float dp_gs = 4.0f * ca_gs * (1.0f - p_gs) - 1.5f * p_gs;

    const float arg   = -k_gs * d_over_kt * (u - 0.25f);  // -k_gs*D/KT*(u - D/2)
    const float p_t0  = 1.0f / (1.0f + e_exp * __expf(arg));
    const float dp_t  = (p_t0 - p_t) * 10.0f;             // / TAU_T

    // ---- stage results, then b128 store LDS -> global ----
    __syncthreads();
    lds[tid * 5 + 0] = dx_hb;
    lds[tid * 5 + 1] = dx_a;
    lds[tid * 5 + 2] = dp_m;
    lds[tid * 5 + 3] = dp_gs;
    lds[tid * 5 + 4] = dp_t;
    __syncthreads();

    v4f* dx4 = (v4f*)dx + blk * VEC4_PER_BLOCK;
#if HAVE_ASYNC_LDS
    // gfx1250 async DMA: LDS -> memory; S_ENDPGM drains ASYNCcnt implicitly.
    ASYNC_L2G_B128(&dx4[tid], &lds4[tid]);
    if (tid < VEC4_PER_BLOCK - ROWS_PER_BLOCK)
        ASYNC_L2G_B128(&dx4[tid + ROWS_PER_BLOCK], &lds4[tid + ROWS_PER_BLOCK]);
#else
    #pragma unroll
    for (int j = 0; j < 2; ++j) {
        int idx = tid + j * ROWS_PER_BLOCK;
        if (idx < VEC4_PER_BLOCK)
            __builtin_nontemporal_store(lds4[idx], &dx4[idx]);
    }
#endif
}

// g[b] = diag(HB_NOISE, A_NOISE, 0, 0): constant 64B pattern, repeated 4M times.
// One float4 per thread, unit stride, NT stores — pure 256 MB write stream.
__global__ __launch_bounds__(256)
void hairbundle_g_kernel(v4f* __restrict__ g, float hb_noise, float a_noise)
{
    const unsigned long long i =
        (unsigned long long)blockIdx.x * 256ull + threadIdx.x;  // float4 index
    v4f v = (v4f){0.0f, 0.0f, 0.0f, 0.0f};
    const unsigned m = (unsigned)i & 3u;   // position within 16-float row
    if (m == 0u)      v.x = hb_noise;      // g[b][0][0]
    else if (m == 1u) v.y = a_noise;       // g[b][1][1]
    __builtin_nontemporal_store(v, &g[i]);
}

extern "C" void kernel_launch(void* const* d_in, const int* in_sizes, int n_in,
                              void* d_out, int out_size, void* d_ws, size_t ws_size,
                              hipStream_t stream)
{
    (void)in_sizes; (void)n_in; (void)out_size; (void)d_ws; (void)ws_size;

    const float* t_ptr = (const float*)d_in[0];   // scalar t (device)
    const float* x     = (const float*)d_in[1];   // (B,5) float32
    float* dx_out = (float*)d_out;                               // B*5 floats
    v4f*   g_out  = (v4f*)((float*)d_out + (size_t)B_TOTAL * 5); // B*16 floats (16B aligned)

    // ---- fold physics constants in double precision on host ----
    const double K_B      = 1.380649e-23;
    const double E_CHG    = 1.602176634e-19;
    const double FARADAY  = 96485.33212331001;
    const double TEMP     = 7.242970516e+22;
    const double Z_CA     = 2.0;
    const double D_CA     = 1e+18;
    const double R_M      = 1.55e+18;
    const double R_GS     = 3.1e+18;
    const double V_M      = -3.120755e+18;
    const double E_T_CA   = 0.0;
    const double P_T_CA   = 1e13;
    const double PI       = 3.14159265358979323846;

    const double KT        = K_B * TEMP;
    const double ARG0      = Z_CA * E_CHG * V_M / KT;
    const double G_T_CA_MAX = P_T_CA * Z_CA * Z_CA * E_CHG * FARADAY / KT
                              * (2.0 - 1.0) / (1.0 - std::exp(ARG0));
    const double DELTA_V   = V_M - E_T_CA;
    const double CA2_DENOM = 2.0 * PI * Z_CA * E_CHG * D_CA;

    const float cam_coef  = (float)(-G_T_CA_MAX * DELTA_V / (CA2_DENOM * R_M));
    const float cags_coef = (float)(-G_T_CA_MAX * DELTA_V / (CA2_DENOM * R_GS));
    const float d_over_kt = (float)(0.5 / KT);            // D / KT
    const float e_exp     = (float)std::exp(0.5 / KT);    // exp(DELTA_E/KT)
    const float hb_noise  = (float)std::sqrt(2.0 * KT);   // EPSILON*sqrt(2KT/LAMBDA_HB)
    const float a_noise   = (float)std::sqrt(2.0 * KT);
    const float t_scale   = (float)(1.0 / 2000000.0);     // OMEGA_0 / HALF

    // 4,000,000 rows / 256 = 15625 blocks exactly
    hairbundle_dx_kernel<<<B_TOTAL / ROWS_PER_BLOCK, ROWS_PER_BLOCK, 0, stream>>>(
        t_ptr, x, dx_out, cam_coef, cags_coef, d_over_kt, e_exp, t_scale);

    // 64,000,000 g floats = 16,000,000 float4 = 62500 blocks * 256 threads exactly
    hairbundle_g_kernel<<<(B_TOTAL * 16 / 4) / 256, 256, 0, stream>>>(
        g_out, hb_noise, a_noise);
}